// CoAttention_61821759259427
// MI455X (gfx1250) — compile-verified
//
#include <hip/hip_runtime.h>
#include <hip/hip_bf16.h>

typedef float v2f __attribute__((ext_vector_type(2)));
typedef float v4f __attribute__((ext_vector_type(4)));
typedef float v8f __attribute__((ext_vector_type(8)));

#define B_   32
#define LQ_  512
#define LK_  2048
#define D_   768
#define D4_  (D_ / 4)          // 192 float4 per row

// ---------------------------------------------------------------------------
// 1a) partial row-sum of sentence_rep over q: partial[b][qc][d], qc in [0,16)
// grid 32*16, block 192 (one float4 column per thread; b128 coalesced rows)
// ---------------------------------------------------------------------------
__global__ void k_partial_sum(const float* __restrict__ S, float* __restrict__ partial) {
    int b = blockIdx.x >> 4, qc = blockIdx.x & 15;
    const v4f* src = (const v4f*)(S + ((size_t)b * LQ_ + (size_t)qc * 32) * D_);
    v4f acc = {};
    for (int q = 0; q < 32; ++q) {
        __builtin_prefetch((const float*)(src + (size_t)(q + 1) * D4_), 0, 1);
        acc += src[(size_t)q * D4_ + threadIdx.x];
    }
    ((v4f*)(partial + ((size_t)b * 16 + qc) * D_))[threadIdx.x] = acc;
}

// ---------------------------------------------------------------------------
// 1b) reduce 16 partials -> [32 x 768]. grid 24, block 256 (float4 lanes).
// ---------------------------------------------------------------------------
__global__ void k_reduce16(const float* __restrict__ partial, float* __restrict__ dst) {
    int i = blockIdx.x * 256 + threadIdx.x;          // i < 32*192
    int b = i / D4_, d4 = i % D4_;
    v4f s = {};
    #pragma unroll
    for (int j = 0; j < 16; ++j)
        s += ((const v4f*)(partial + ((size_t)b * 16 + j) * D_))[d4];
    ((v4f*)dst)[i] = s;
}

// ---------------------------------------------------------------------------
// WMMA f32 GEMM: Dst[32 x 768] = A[32 x 768] * B + biasScale*bias
//   TRANSB=1: B(k,n) = W[n*768 + k]   (A @ W^T);  TRANSB=0: B(k,n) = W[k*768+n]
// 96 waves (2 M-tiles x 48 N-tiles): grid 12, block 256. EXEC all ones.
// V_WMMA_F32_16X16X4_F32 layouts per CDNA5 ISA 7.12.2.
// ---------------------------------------------------------------------------
template <bool TRANSB, bool HASBIAS>
__global__ void k_gemm_wmma(const float* __restrict__ A, const float* __restrict__ W,
                            const float* __restrict__ bias, float biasScale,
                            float* __restrict__ Dst) {
    int wave = (blockIdx.x * 256 + threadIdx.x) >> 5;   // 0..95
    int lane = threadIdx.x & 31;
    int mt = wave & 1, nt = wave >> 1;                  // 2 x 48 tiles
    int m0 = mt * 16, n0 = nt * 16;
    int row = lane & 15;                                // M (A) / N (B,C,D)
    int kg  = lane >> 4;                                // K-pair select

    v8f c = {};
    for (int k0 = 0; k0 < D_; k0 += 4) {
        int k = k0 + 2 * kg;
        v2f a = *(const v2f*)(A + (size_t)(m0 + row) * D_ + k);   // K=k, K=k+1
        v2f b;
        if (TRANSB) {
            b = *(const v2f*)(W + (size_t)(n0 + row) * D_ + k);
        } else {
            b.x = W[(size_t)(k    ) * D_ + n0 + row];
            b.y = W[(size_t)(k + 1) * D_ + n0 + row];
        }
        c = __builtin_amdgcn_wmma_f32_16x16x4_f32(
                /*neg_a=*/false, a, /*neg_b=*/false, b,
                /*c_mod=*/(short)0, c, /*reuse_a=*/false, /*reuse_b=*/false);
    }
    float bv = HASBIAS ? (biasScale * bias[n0 + row]) : 0.f;
    #pragma unroll
    for (int j = 0; j < 8; ++j) {                        // VGPR j: M=j (+8 for hi lanes)
        int m = m0 + j + 8 * kg;
        Dst[(size_t)m * D_ + n0 + row] = c[j] + bv;
    }
}

// ---------------------------------------------------------------------------
// scores[b,k] = comment[b,k,:] . t[b,:] + q_sum[b,:] . bk
// grid 32*8 (chunks of 256 k-rows), block 256 = 8 waves, wave-per-row,
// b128 loads (wave reads 512B contiguous per step).
// ---------------------------------------------------------------------------
__global__ void k_scores(const float* __restrict__ Cm, const float* __restrict__ t,
                         const float* __restrict__ q_sum, const float* __restrict__ bk,
                         float* __restrict__ scores) {
    __shared__ v4f tl4[D4_];
    int b = blockIdx.x >> 3, kc = blockIdx.x & 7;
    if (threadIdx.x < D4_)
        tl4[threadIdx.x] = ((const v4f*)(t + (size_t)b * D_))[threadIdx.x];
    __syncthreads();

    int lane = threadIdx.x & 31, wave = threadIdx.x >> 5;

    // cpart = q_sum[b] . bk   (redundant per wave; tiny, L2-hit)
    v4f ca = {};
    #pragma unroll
    for (int j = 0; j < D4_ / 32; ++j)
        ca += ((const v4f*)(q_sum + (size_t)b * D_))[lane + 32 * j] *
              ((const v4f*)bk)[lane + 32 * j];
    float cpart = ca.x + ca.y + ca.z + ca.w;
    #pragma unroll
    for (int off = 16; off; off >>= 1) cpart += __shfl_xor(cpart, off, 32);

    int kbase = kc * 256 + wave * 32;
    for (int r = 0; r < 32; ++r) {
        const v4f* rowp = (const v4f*)(Cm + ((size_t)b * LK_ + kbase + r) * D_);
        __builtin_prefetch((const float*)(rowp + D4_), 0, 1);  // next row
        v4f pa = {};
        #pragma unroll
        for (int j = 0; j < D4_ / 32; ++j)
            pa += rowp[lane + 32 * j] * tl4[lane + 32 * j];
        float p = pa.x + pa.y + pa.z + pa.w;
        #pragma unroll
        for (int off = 16; off; off >>= 1) p += __shfl_xor(p, off, 32);
        if (lane == 0) scores[(size_t)b * LK_ + kbase + r] = p + cpart;
    }
}

// ---------------------------------------------------------------------------
// per-batch softmax over 2048 scores. grid 32, block 256 (2 float4 / thread).
// ---------------------------------------------------------------------------
__global__ void k_softmax(const float* __restrict__ scores, float* __restrict__ alpha) {
    __shared__ float red[8];
    int b = blockIdx.x, lane = threadIdx.x & 31, wave = threadIdx.x >> 5;
    const v4f* s4 = (const v4f*)(scores + (size_t)b * LK_);
    v4f v[2];
    #pragma unroll
    for (int j = 0; j < 2; ++j) v[j] = s4[threadIdx.x + j * 256];

    float m = -3.0e38f;
    #pragma unroll
    for (int j = 0; j < 2; ++j)
        m = fmaxf(fmaxf(fmaxf(m, v[j].x), fmaxf(v[j].y, v[j].z)), v[j].w);
    #pragma unroll
    for (int off = 16; off; off >>= 1) m = fmaxf(m, __shfl_xor(m, off, 32));
    if (lane == 0) red[wave] = m;
    __syncthreads();
    float bm = red[0];
    #pragma unroll
    for (int w = 1; w < 8; ++w) bm = fmaxf(bm, red[w]);
    __syncthreads();

    float s = 0.f;
    #pragma unroll
    for (int j = 0; j < 2; ++j) {
        v[j].x = __expf(v[j].x - bm); v[j].y = __expf(v[j].y - bm);
        v[j].z = __expf(v[j].z - bm); v[j].w = __expf(v[j].w - bm);
        s += v[j].x + v[j].y + v[j].z + v[j].w;
    }
    #pragma unroll
    for (int off = 16; off; off >>= 1) s += __shfl_xor(s, off, 32);
    if (lane == 0) red[wave] = s;
    __syncthreads();
    float bs = 0.f;
    #pragma unroll
    for (int w = 0; w < 8; ++w) bs += red[w];
    float inv = 1.0f / bs;
    v4f* a4 = (v4f*)(alpha + (size_t)b * LK_);
    #pragma unroll
    for (int j = 0; j < 2; ++j) a4[threadIdx.x + j * 256] = v[j] * inv;
}

// ---------------------------------------------------------------------------
// vpart[b][kc][d] = sum_{k in chunk of 128} alpha[b,k] * comment[b,k,d]
// grid 32*16, block 192 (one float4 column per thread; b128 coalesced rows).
// ---------------------------------------------------------------------------
__global__ void k_vctx_partial(const float* __restrict__ Cm, const float* __restrict__ alpha,
                               float* __restrict__ vpart) {
    __shared__ float al[128];
    int b = blockIdx.x >> 4, kc = blockIdx.x & 15;
    if (threadIdx.x < 128)
        al[threadIdx.x] = alpha[(size_t)b * LK_ + kc * 128 + threadIdx.x];
    __syncthreads();
    const v4f* base = (const v4f*)(Cm + ((size_t)b * LK_ + (size_t)kc * 128) * D_);
    v4f acc = {};
    for (int k = 0; k < 128; ++k) {
        __builtin_prefetch((const float*)(base + (size_t)(k + 1) * D4_), 0, 1);
        acc += al[k] * base[(size_t)k * D4_ + threadIdx.x];
    }
    ((v4f*)(vpart + ((size_t)b * 16 + kc) * D_))[threadIdx.x] = acc;
}

// ---------------------------------------------------------------------------
extern "C" void kernel_launch(void* const* d_in, const int* in_sizes, int n_in,
                              void* d_out, int out_size, void* d_ws, size_t ws_size,
                              hipStream_t stream) {
    const float* S  = (const float*)d_in[0];   // sentence_rep [32,512,768]
    const float* Cm = (const float*)d_in[1];   // comment_rep  [32,2048,768]
    const float* Wq = (const float*)d_in[2];
    const float* bq = (const float*)d_in[3];
    const float* Wk = (const float*)d_in[4];
    const float* bk = (const float*)d_in[5];
    const float* Wv = (const float*)d_in[6];
    const float* bv = (const float*)d_in[7];
    float* out = (float*)d_out;

    float* ws = (float*)d_ws;
    float* w_partial = ws;                        // 32*16*768 (reused for vpart)
    float* w_ssum    = w_partial + 32 * 16 * D_;  // 32*768
    float* w_qsum    = w_ssum  + B_ * D_;
    float* w_t       = w_qsum  + B_ * D_;
    float* w_scores  = w_t     + B_ * D_;
    float* w_alpha   = w_scores + B_ * LK_;
    float* w_vctx    = w_alpha  + B_ * LK_;

    // 1) s_sum = sum_q sentence_rep
    k_partial_sum<<<dim3(B_ * 16), dim3(192), 0, stream>>>(S, w_partial);
    k_reduce16  <<<dim3(B_ * D4_ / 256), dim3(256), 0, stream>>>(w_partial, w_ssum);

    // 2) q_sum = s_sum @ Wq^T + 512*bq        (WMMA)
    k_gemm_wmma<true, true><<<dim3(12), dim3(256), 0, stream>>>(w_ssum, Wq, bq, (float)LQ_, w_qsum);

    // 3) t = q_sum @ Wk                       (WMMA)
    k_gemm_wmma<false, false><<<dim3(12), dim3(256), 0, stream>>>(w_qsum, Wk, nullptr, 0.f, w_t);

    // 4) scores[b,k] = comment . t + q_sum . bk
    k_scores<<<dim3(B_ * 8), dim3(256), 0, stream>>>(Cm, w_t, w_qsum, bk, w_scores);

    // 5) alpha = softmax(scores)
    k_softmax<<<dim3(B_), dim3(256), 0, stream>>>(w_scores, w_alpha);

    // 6) v_ctx = sum_k alpha * comment
    k_vctx_partial<<<dim3(B_ * 16), dim3(192), 0, stream>>>(Cm, w_alpha, w_partial);
    k_reduce16   <<<dim3(B_ * D4_ / 256), dim3(256), 0, stream>>>(w_partial, w_vctx);

    // 7) out = v_ctx @ Wv^T + bv              (WMMA)
    k_gemm_wmma<true, true><<<dim3(12), dim3(256), 0, stream>>>(w_vctx, Wv, bv, 1.0f, out);
}